// HdcGenericEncoder_90692529422505
// MI455X (gfx1250) — compile-verified
//
#include <hip/hip_runtime.h>

// Problem constants (from reference: B=64, T=4, F=64, D=10000, NGRAMS=7)
#define D_DIM   10000
#define B_DIM   64
#define F_DIM   64
#define TF      256            // T*F, stride between batches in x; t=0 row is first F floats
#define NG      7
#define W_DIM   57             // F - NGRAMS
#define TILE    256            // d's per block
#define CSEG    (TILE + NG - 1) // 262 staged columns per row (need d-6 .. d0+TILE-1)
#define PITCH   264            // LDS row pitch in floats
#define NROWS   63             // rows f = 1..63 participate in the windows

#if defined(__AMDGCN__) && __has_builtin(__builtin_amdgcn_global_load_async_to_lds_b32)
#define HAS_ASYNC_LDS 1
typedef __attribute__((address_space(1))) int* as1_int_ptr;
typedef __attribute__((address_space(3))) int* as3_int_ptr;
#else
#define HAS_ASYNC_LDS 0
#endif

__global__ __launch_bounds__(256) void hdc_encoder_kernel(
    const float* __restrict__ x,       // (B, T, F)
    const float* __restrict__ keys,    // (F, D), rows 0..56 used
    const float* __restrict__ motion,  // (3000, D), +/-1
    const float* __restrict__ hr,      // (200, D), +/-1
    float* __restrict__ out)           // (B, D)
{
    __shared__ float seg[NROWS * PITCH];
    __shared__ const float* rowptr[F_DIM];

    const int b   = blockIdx.y;
    const int d0  = blockIdx.x * TILE;
    const int t   = threadIdx.x;
    const int rem = (D_DIM - d0 < TILE) ? (D_DIM - d0) : TILE;

    // ---- 1) data-dependent row pointers (level lookup), threads 0..63 ----
    if (t < F_DIM) {
        const float v = x[b * TF + t];           // x[b, 0, t]
        const float* rp;
        if (t < F_DIM - 1) {
            // motion: round((v - (-3)) / 6 * 2999), RNE like jnp.round, clamp after
            float u = (v + 3.0f) / 6.0f * 2999.0f;
            int id = (int)__builtin_rintf(u);
            id = id < 0 ? 0 : (id > 2999 ? 2999 : id);
            rp = motion + (long)id * D_DIM;
        } else {
            // hr: round((v - 50) / 150 * 199)
            float u = (v - 50.0f) / 150.0f * 199.0f;
            int id = (int)__builtin_rintf(u);
            id = id < 0 ? 0 : (id > 199 ? 199 : id);
            rp = hr + (long)id * D_DIM;
        }
        rowptr[t] = rp;
    }
    __syncthreads();

    // ---- 2) stage 63 row segments [d0-6, d0+rem) (mod D) into LDS ----
    // CDNA5 path: async global->LDS DMA tracked by ASYNCcnt.
    for (int s = t; s < NROWS * CSEG; s += 256) {
        const int fi = s / CSEG;          // 0..62  -> f = fi+1
        const int c  = s - fi * CSEG;     // 0..261
        int dd = d0 + c - (NG - 1);
        if (dd < 0) dd += D_DIM;          // wrap only happens for tile 0
        if (dd < D_DIM) {                 // skip past-the-end cols of the last tile
            const float* gp = rowptr[fi + 1] + dd;
            float* lp = &seg[fi * PITCH + c];
#if HAS_ASYNC_LDS
            __builtin_amdgcn_global_load_async_to_lds_b32(
                (as1_int_ptr)gp, (as3_int_ptr)lp, 0, 0);
#else
            *lp = *gp;
#endif
        }
    }
#if HAS_ASYNC_LDS
    __builtin_amdgcn_s_wait_asynccnt(0);
#endif
    __syncthreads();

    // ---- 3) 57 sliding 7-gram products + keyed accumulation ----
    if (t < rem) {
        const int d = d0 + t;
        float acc = 0.0f;
#pragma unroll 3
        for (int w = 0; w < W_DIM; ++w) {
            // window w, factor j: enc row f=w+1+j at D-offset d-6+j  ->  seg[(w+j)][t+j]
            const float* sp = &seg[w * PITCH + t];
            float p = sp[0];
            p *= sp[1 * PITCH + 1];
            p *= sp[2 * PITCH + 2];
            p *= sp[3 * PITCH + 3];
            p *= sp[4 * PITCH + 4];
            p *= sp[5 * PITCH + 5];
            p *= sp[6 * PITCH + 6];
            acc = __builtin_fmaf(keys[w * D_DIM + d], p, acc);
        }
        out[b * D_DIM + d] = acc;
    }
}

extern "C" void kernel_launch(void* const* d_in, const int* in_sizes, int n_in,
                              void* d_out, int out_size, void* d_ws, size_t ws_size,
                              hipStream_t stream) {
    const float* x      = (const float*)d_in[0];  // (64, 4, 64)
    const float* keys   = (const float*)d_in[1];  // (64, 10000)
    const float* motion = (const float*)d_in[2];  // (3000, 10000)
    const float* hr     = (const float*)d_in[3];  // (200, 10000)
    float* out = (float*)d_out;                   // (64, 10000)

    dim3 grid((D_DIM + TILE - 1) / TILE, B_DIM);  // (40, 64)
    hdc_encoder_kernel<<<grid, dim3(256), 0, stream>>>(x, keys, motion, hr, out);
}